// VQEGNN_781684048212
// MI455X (gfx1250) — compile-verified
//
#include <hip/hip_runtime.h>
#include <hip/hip_bf16.h>

// ---------------------------------------------------------------------------
// VQ-EGNN forward for MI455X (gfx1250), fp32 WMMA (V_WMMA_F32_16X16X4_F32).
// GEMM v2: block tile 128 x NC (4 wave32 x 32 rows), W staged in LDS in
// fragment-packed order so each B fragment is one aligned ds_load_b64, and
// each B fragment feeds two WMMAs (two M-subtiles per wave).
// ---------------------------------------------------------------------------

typedef __attribute__((ext_vector_type(2))) float v2f;
typedef __attribute__((ext_vector_type(8))) float v8f;

#define N_NODES 20000
#define N_EDGES 160000
#define HID     128
#define LAT     64
#define N_CODES 512
#define COMMIT  0.25f
#define LN_EPS  1e-5f

__device__ __forceinline__ float silu_f(float x) { return x / (1.0f + __expf(-x)); }

// ---------------------------------------------------------------------------
// WMMA GEMM:  C[M,NC] = act( concat(A0[idx0] , A1[idx1] , tail) @ W + bias ) (+resid)
//   A0 rows have width K0, A1 width K1, tail width Kt (K = K0+K1+Kt).
//   idx0/idx1 == nullptr -> identity row index (no gather).
//   Block: 128 threads (4 wave32). Tile: 128 rows x NC cols; each wave owns
//   32 rows = two 16-row WMMA subtiles. K chunked by 32.
//   Fragment layouts per CDNA5 ISA 7.12.2 (f32 16x16x4):
//     A: lane m (and m+16) holds row m's K-pair (k,k+1) / (k+2,k+3)
//     B: lane n (and n+16) holds col n's K-pair rows (k,k+1) / (k+2,k+3)
//   Bs is stored fragment-packed: group g = k>>1 holds [B[2g][n], B[2g+1][n]]
//   contiguously per column -> one aligned b64 LDS read per fragment.
// ---------------------------------------------------------------------------
template <int NT, int ACT>
__global__ __launch_bounds__(128) void gemm_kernel(
    const float* __restrict__ A0, const int* __restrict__ idx0, int K0,
    const float* __restrict__ A1, const int* __restrict__ idx1, int K1,
    const float* __restrict__ tail, int Kt,
    const float* __restrict__ W, const float* __restrict__ bias,
    const float* __restrict__ resid, float* __restrict__ C, int M)
{
  constexpr int NC = NT * 16;
  constexpr int GS = NC * 2 + 32;          // packed-B group stride (bank swizzle)
  const int K = K0 + K1 + Kt;

  __shared__ __align__(16) float As[128 * 36];
  __shared__ __align__(16) float Bs[16 * GS];

  const int tid   = threadIdx.x;
  const int lane  = tid & 31;
  const int wv    = tid >> 5;
  const int ln15  = lane & 15;
  const int khalf = (lane >> 4) << 1;      // 0 for lanes 0-15, 2 for lanes 16-31
  const int base_m = blockIdx.x * 128;
  const int mrow0 = wv * 32 + ln15;        // first M-subtile row for this lane

  v8f acc0[NT], acc1[NT];
#pragma unroll
  for (int i = 0; i < NT; ++i) {
    acc0[i] = v8f{0.f, 0.f, 0.f, 0.f, 0.f, 0.f, 0.f, 0.f};
    acc1[i] = v8f{0.f, 0.f, 0.f, 0.f, 0.f, 0.f, 0.f, 0.f};
  }

  for (int kc = 0; kc < K; kc += 32) {
    // ---- stage A chunk (128 x 32) with segment gather/concat ----
    for (int e = tid; e < 128 * 32; e += 128) {
      int r = e >> 5, c = e & 31;
      int row = base_m + r, col = kc + c;
      float v = 0.f;
      if (row < M && col < K) {
        if (col < K0) {
          int g = idx0 ? idx0[row] : row;
          v = A0[(size_t)g * K0 + col];
        } else if (col < K0 + K1) {
          int g = idx1 ? idx1[row] : row;
          v = A1[(size_t)g * K1 + (col - K0)];
        } else {
          v = tail[(size_t)row * Kt + (col - K0 - K1)];
        }
      }
      As[r * 36 + c] = v;
    }
    // ---- stage W chunk (32 x NC), fragment-packed ----
    for (int e = tid; e < 32 * NC; e += 128) {
      int r = e / NC, c = e % NC;
      int krow = kc + r;
      float v = (krow < K) ? W[(size_t)krow * NC + c] : 0.f;
      Bs[(r >> 1) * GS + c * 2 + (r & 1)] = v;
    }
    __syncthreads();

    // ---- 8 K-steps of f32 WMMA per chunk; each B fragment feeds 2 WMMAs ----
#pragma unroll
    for (int kk = 0; kk < 32; kk += 4) {
      v2f a0 = *(const v2f*)&As[(mrow0)      * 36 + kk + khalf];
      v2f a1 = *(const v2f*)&As[(mrow0 + 16) * 36 + kk + khalf];
      const float* bg = &Bs[((kk + khalf) >> 1) * GS];
#pragma unroll
      for (int nt = 0; nt < NT; ++nt) {
        v2f b = *(const v2f*)&bg[(nt * 16 + ln15) * 2];
        acc0[nt] = __builtin_amdgcn_wmma_f32_16x16x4_f32(
            false, a0, false, b, (short)0, acc0[nt], false, false);
        acc1[nt] = __builtin_amdgcn_wmma_f32_16x16x4_f32(
            false, a1, false, b, (short)0, acc1[nt], false, false);
      }
    }
    __syncthreads();
  }

  // ---- epilogue: bias + activation + optional residual ----
#pragma unroll
  for (int mt = 0; mt < 2; ++mt) {
#pragma unroll
    for (int nt = 0; nt < NT; ++nt) {
#pragma unroll
      for (int v = 0; v < 8; ++v) {
        int row = base_m + wv * 32 + mt * 16 + v + ((lane >> 4) << 3);
        int col = nt * 16 + ln15;
        if (row < M) {
          float x = (mt ? acc1[nt][v] : acc0[nt][v]) + bias[col];
          if (ACT == 1) x = silu_f(x);
          if (resid) x += resid[(size_t)row * NC + col];
          C[(size_t)row * NC + col] = x;
        }
      }
    }
  }
}

// ---------------------------------------------------------------------------
__global__ void embed_kernel(const int* __restrict__ seq,
                             const float* __restrict__ emb,
                             float* __restrict__ h0)
{
  int id = blockIdx.x * blockDim.x + threadIdx.x;
  if (id >= N_NODES * 32) return;
  h0[id] = emb[seq[id >> 5] * 32 + (id & 31)];
}

__global__ void edge_geom_kernel(const float* __restrict__ coords,
                                 const int* __restrict__ src,
                                 const int* __restrict__ dst,
                                 const float* __restrict__ eattr,
                                 float* __restrict__ tail,
                                 float* __restrict__ cdiff)
{
  int e = blockIdx.x * blockDim.x + threadIdx.x;
  if (e >= N_EDGES) return;
  int s = src[e], d = dst[e];
  float dx = coords[s * 3 + 0] - coords[d * 3 + 0];
  float dy = coords[s * 3 + 1] - coords[d * 3 + 1];
  float dz = coords[s * 3 + 2] - coords[d * 3 + 2];
  cdiff[e * 3 + 0] = dx;
  cdiff[e * 3 + 1] = dy;
  cdiff[e * 3 + 2] = dz;
  tail[e * 2 + 0] = dx * dx + dy * dy + dz * dz;
  tail[e * 2 + 1] = eattr[e];
}

// cw = silu(m@c1+b1) @ c2 (the silu'd GEMM result is `t`); scatter coord update.
__global__ void coord_scatter_kernel(const float* __restrict__ t,
                                     const float* __restrict__ c2,
                                     const float* __restrict__ cdiff,
                                     const int* __restrict__ dst,
                                     float* __restrict__ aggc,
                                     float* __restrict__ cnt)
{
  int e = blockIdx.x * blockDim.x + threadIdx.x;
  if (e >= N_EDGES) return;
  const float4* tr  = (const float4*)(t + (size_t)e * HID);
  const float4* c2v = (const float4*)c2;
  float cw = 0.f;
#pragma unroll 8
  for (int j = 0; j < HID / 4; ++j) {
    float4 a = tr[j], b = c2v[j];
    cw += a.x * b.x + a.y * b.y + a.z * b.z + a.w * b.w;
  }
  int d = dst[e];
  unsafeAtomicAdd(&aggc[d * 3 + 0], cdiff[e * 3 + 0] * cw);
  unsafeAtomicAdd(&aggc[d * 3 + 1], cdiff[e * 3 + 1] * cw);
  unsafeAtomicAdd(&aggc[d * 3 + 2], cdiff[e * 3 + 2] * cw);
  unsafeAtomicAdd(&cnt[d], 1.0f);
}

__global__ void feat_scatter_kernel(const float* __restrict__ m,
                                    const int* __restrict__ dst,
                                    float* __restrict__ agg)
{
  size_t id = (size_t)blockIdx.x * blockDim.x + threadIdx.x;
  if (id >= (size_t)N_EDGES * HID) return;
  int e = (int)(id >> 7), j = (int)(id & 127);
  unsafeAtomicAdd(&agg[(size_t)dst[e] * HID + j], m[id]);
}

__global__ void coords_update_kernel(float* __restrict__ coords,
                                     const float* __restrict__ aggc,
                                     const float* __restrict__ cnt)
{
  int id = blockIdx.x * blockDim.x + threadIdx.x;
  if (id >= N_NODES * 3) return;
  int i = id / 3;
  float c = cnt[i];
  float s = c > 1.0f ? c : 1.0f;
  coords[id] += aggc[id] / s;
}

__global__ void layernorm_kernel(float* __restrict__ h)
{
  int i = blockIdx.x * blockDim.x + threadIdx.x;
  if (i >= N_NODES) return;
  float4* p = (float4*)(h + (size_t)i * LAT);
  float4 r[LAT / 4];
  float s = 0.f;
#pragma unroll
  for (int j = 0; j < LAT / 4; ++j) {
    r[j] = p[j];
    s += r[j].x + r[j].y + r[j].z + r[j].w;
  }
  float mu = s * (1.0f / LAT);
  float vs = 0.f;
#pragma unroll
  for (int j = 0; j < LAT / 4; ++j) {
    float a = r[j].x - mu, b = r[j].y - mu, c = r[j].z - mu, d = r[j].w - mu;
    vs += a * a + b * b + c * c + d * d;
  }
  float inv = rsqrtf(vs * (1.0f / LAT) + LN_EPS);
#pragma unroll
  for (int j = 0; j < LAT / 4; ++j) {
    r[j].x = (r[j].x - mu) * inv;
    r[j].y = (r[j].y - mu) * inv;
    r[j].z = (r[j].z - mu) * inv;
    r[j].w = (r[j].w - mu) * inv;
    p[j] = r[j];
  }
}

// Nearest code (squared L2), straight-through hq = quant, loss accumulation.
__global__ void vq_kernel(const float* __restrict__ h,
                          const float* __restrict__ cb,
                          float* __restrict__ hq,
                          float* __restrict__ loss)
{
  int i = blockIdx.x * blockDim.x + threadIdx.x;
  if (i >= N_NODES) return;
  const float4* hp = (const float4*)(h + (size_t)i * LAT);
  float4 hr[LAT / 4];
#pragma unroll
  for (int j = 0; j < LAT / 4; ++j) hr[j] = hp[j];
  float best = 3.4e38f;
  int bi = 0;
  for (int c = 0; c < N_CODES; ++c) {
    const float4* cp = (const float4*)(cb + (size_t)c * LAT);
    float d = 0.f;
#pragma unroll
    for (int j = 0; j < LAT / 4; ++j) {
      float4 cv = cp[j];
      float a = hr[j].x - cv.x, b = hr[j].y - cv.y,
            e = hr[j].z - cv.z, f = hr[j].w - cv.w;
      d += a * a + b * b + e * e + f * f;
    }
    if (d < best) { best = d; bi = c; }
  }
  const float4* qp = (const float4*)(cb + (size_t)bi * LAT);
  float4* op = (float4*)(hq + (size_t)i * LAT);
#pragma unroll
  for (int j = 0; j < LAT / 4; ++j) op[j] = qp[j];
  unsafeAtomicAdd(loss, best);
}

__global__ void finalize_kernel(const float* __restrict__ loss,
                                float* __restrict__ out2)
{
  float mean = loss[0] * (1.0f / ((float)N_NODES * (float)LAT));
  out2[0] = COMMIT * mean;  // commitment_loss
  out2[1] = mean;           // codebook_loss
}

// ---------------------------------------------------------------------------
// Host-side orchestration
// ---------------------------------------------------------------------------
struct LayerW {
  const float *c1W, *c1b, *c2, *e1W, *e1b, *e2W, *e2b, *n1W, *n1b, *n2W, *n2b;
};
struct EgnnW {
  LayerW L[4];
  const float *winW, *winb, *woutW, *woutb;
};

// jax.tree_util flattens dicts sorted by key:
//  egnn: 'layers' < 'win' < 'wout'; layer: c1 < c2 < e1 < e2 < n1 < n2
static void parse_egnn(void* const* d_in, int& cur, EgnnW& P)
{
  for (int i = 0; i < 4; ++i) {
    LayerW& L = P.L[i];
    L.c1W = (const float*)d_in[cur++]; L.c1b = (const float*)d_in[cur++];
    L.c2  = (const float*)d_in[cur++];
    L.e1W = (const float*)d_in[cur++]; L.e1b = (const float*)d_in[cur++];
    L.e2W = (const float*)d_in[cur++]; L.e2b = (const float*)d_in[cur++];
    L.n1W = (const float*)d_in[cur++]; L.n1b = (const float*)d_in[cur++];
    L.n2W = (const float*)d_in[cur++]; L.n2b = (const float*)d_in[cur++];
  }
  P.winW  = (const float*)d_in[cur++]; P.winb  = (const float*)d_in[cur++];
  P.woutW = (const float*)d_in[cur++]; P.woutb = (const float*)d_in[cur++];
}

static void run_egnn(const EgnnW& P, int Cin, const float* h_in,
                     float* h, float* hb, float* m, float* t,
                     float* tail, float* cdiff,
                     float* agg, float* aggc, float* cnt, float* coords,
                     const int* src, const int* dst, const float* eattr,
                     hipStream_t stream)
{
  dim3 blk(128);
  dim3 gN((N_NODES + 127) / 128), gE((N_EDGES + 127) / 128);

  // win: h = h_in @ Win + b
  gemm_kernel<8, 0><<<gN, blk, 0, stream>>>(
      h_in, nullptr, Cin, nullptr, nullptr, 0, nullptr, 0,
      P.winW, P.winb, nullptr, h, N_NODES);

  for (int l = 0; l < 4; ++l) {
    const LayerW& L = P.L[l];
    edge_geom_kernel<<<(N_EDGES + 255) / 256, 256, 0, stream>>>(
        coords, src, dst, eattr, tail, cdiff);
    // e1: [E,258] (gather-concat) @ [258,128], silu -> t
    gemm_kernel<8, 1><<<gE, blk, 0, stream>>>(
        h, dst, HID, h, src, HID, tail, 2,
        L.e1W, L.e1b, nullptr, t, N_EDGES);
    // e2: silu -> m
    gemm_kernel<8, 1><<<gE, blk, 0, stream>>>(
        t, nullptr, HID, nullptr, nullptr, 0, nullptr, 0,
        L.e2W, L.e2b, nullptr, m, N_EDGES);
    // c1: silu -> t
    gemm_kernel<8, 1><<<gE, blk, 0, stream>>>(
        m, nullptr, HID, nullptr, nullptr, 0, nullptr, 0,
        L.c1W, L.c1b, nullptr, t, N_EDGES);
    // zero agg_feat / agg_coord / cnt (contiguous N*132 floats)
    hipMemsetAsync(agg, 0, (size_t)N_NODES * (HID + 4) * sizeof(float), stream);
    coord_scatter_kernel<<<(N_EDGES + 127) / 128, 128, 0, stream>>>(
        t, L.c2, cdiff, dst, aggc, cnt);
    feat_scatter_kernel<<<(unsigned)(((size_t)N_EDGES * HID + 255) / 256), 256, 0, stream>>>(
        m, dst, agg);
    // n1: concat(h, agg) @ [256,128], silu -> hb
    gemm_kernel<8, 1><<<gN, blk, 0, stream>>>(
        h, nullptr, HID, agg, nullptr, HID, nullptr, 0,
        L.n1W, L.n1b, nullptr, hb, N_NODES);
    // n2: hb @ [128,128] + residual(h) -> h
    gemm_kernel<8, 0><<<gN, blk, 0, stream>>>(
        hb, nullptr, HID, nullptr, nullptr, 0, nullptr, 0,
        L.n2W, L.n2b, h, h, N_NODES);
    coords_update_kernel<<<(N_NODES * 3 + 255) / 256, 256, 0, stream>>>(
        coords, aggc, cnt);
  }
}

extern "C" void kernel_launch(void* const* d_in, const int* in_sizes, int n_in,
                              void* d_out, int out_size, void* d_ws, size_t ws_size,
                              hipStream_t stream)
{
  (void)in_sizes; (void)n_in; (void)out_size; (void)ws_size;

  // Top-level dict insertion order; params leaves in jax sorted-key order.
  const int*   seq      = (const int*)d_in[0];
  const float* pos      = (const float*)d_in[1];
  const int*   ei       = (const int*)d_in[2];
  const int*   src      = ei;
  const int*   dst      = ei + N_EDGES;
  const float* eattr    = (const float*)d_in[3];
  const float* init_pos = (const float*)d_in[4];
  int cur = 5;
  const float* codebook = (const float*)d_in[cur++];
  EgnnW dec, enc;
  parse_egnn(d_in, cur, dec);
  parse_egnn(d_in, cur, enc);
  const float* seq_emb = (const float*)d_in[cur++];

  // ---- workspace partition (1 KB aligned chunks) ----
  float* p = (float*)d_ws;
  auto take = [&](size_t nfl) {
    float* r = p;
    p += (nfl + 255) & ~(size_t)255;
    return r;
  };
  float* h      = take((size_t)N_NODES * HID);
  float* hb     = take((size_t)N_NODES * HID);
  float* agg    = take((size_t)N_NODES * (HID + 4));   // agg | aggc | cnt contiguous
  float* aggc   = agg + (size_t)N_NODES * HID;
  float* cnt    = aggc + (size_t)N_NODES * 3;
  float* coords = take((size_t)N_NODES * 3);
  float* hlat   = take((size_t)N_NODES * LAT);
  float* hq     = take((size_t)N_NODES * LAT);
  float* m      = take((size_t)N_EDGES * HID);
  float* t      = take((size_t)N_EDGES * HID);
  float* tail   = take((size_t)N_EDGES * 2);
  float* cdiff  = take((size_t)N_EDGES * 3);
  float* loss   = take(256);

  // ---- encoder ----
  embed_kernel<<<(N_NODES * 32 + 255) / 256, 256, 0, stream>>>(seq, seq_emb, hb);
  hipMemcpyAsync(coords, pos, (size_t)N_NODES * 3 * sizeof(float),
                 hipMemcpyDeviceToDevice, stream);
  run_egnn(enc, 32, hb, h, hb, m, t, tail, cdiff, agg, aggc, cnt, coords,
           src, dst, eattr, stream);
  // wout: [N,128] @ [128,64] -> hlat
  gemm_kernel<4, 0><<<dim3((N_NODES + 127) / 128), dim3(128), 0, stream>>>(
      h, nullptr, HID, nullptr, nullptr, 0, nullptr, 0,
      enc.woutW, enc.woutb, nullptr, hlat, N_NODES);

  // ---- layernorm + VQ ----
  layernorm_kernel<<<(N_NODES + 255) / 256, 256, 0, stream>>>(hlat);
  hipMemsetAsync(loss, 0, 256 * sizeof(float), stream);
  vq_kernel<<<(N_NODES + 127) / 128, 128, 0, stream>>>(hlat, codebook, hq, loss);

  // ---- decoder (its wout output is unused by the reference; skipped) ----
  hipMemcpyAsync(coords, init_pos, (size_t)N_NODES * 3 * sizeof(float),
                 hipMemcpyDeviceToDevice, stream);
  run_egnn(dec, LAT, hq, h, hb, m, t, tail, cdiff, agg, aggc, cnt, coords,
           src, dst, eattr, stream);

  // ---- outputs: pos_recon [60000] | commitment | codebook ----
  hipMemcpyAsync(d_out, coords, (size_t)N_NODES * 3 * sizeof(float),
                 hipMemcpyDeviceToDevice, stream);
  finalize_kernel<<<1, 1, 0, stream>>>(loss, (float*)d_out + N_NODES * 3);
}